// EAMPotential_84018150244719
// MI455X (gfx1250) — compile-verified
//
#include <hip/hip_runtime.h>
#include <stdint.h>

namespace {
constexpr int kB = 16;
constexpr int kN = 1024;
constexpr int kRowsPerBlock = 8;           // 8 wave32 per block, one row each for staging
constexpr int kBlocksX = kN / kRowsPerBlock; // 128
constexpr float kLog2e = 1.4426950408889634f;
}

// ---------- CDNA5-specific helpers (builtin with inline-asm fallback) ----------

__device__ __forceinline__ float fexp2(float x) {
#if __has_builtin(__builtin_amdgcn_exp2f)
  return __builtin_amdgcn_exp2f(x);   // v_exp_f32
#else
  return exp2f(x);
#endif
}

// Async global -> LDS copy, 16B per lane (GLOBAL_LOAD_ASYNC_TO_LDS_B128, ASYNCcnt)
typedef int v4i_t __attribute__((ext_vector_type(4)));
typedef __attribute__((address_space(1))) v4i_t gv4_t;   // global (prints as __device__)
typedef __attribute__((address_space(3))) v4i_t lv4_t;   // LDS    (prints as __shared__)

__device__ __forceinline__ void async_copy_b128(const void* gsrc, void* ldst) {
#if __has_builtin(__builtin_amdgcn_global_load_async_to_lds_b128)
  __builtin_amdgcn_global_load_async_to_lds_b128(
      (gv4_t*)(uintptr_t)gsrc, (lv4_t*)(uint32_t)(uintptr_t)ldst, 0, 0);
#else
  uint32_t l = (uint32_t)(uintptr_t)ldst;        // low 32 bits of generic addr = LDS offset
  uint64_t g = (uint64_t)(uintptr_t)gsrc;
  asm volatile("global_load_async_to_lds_b128 %0, %1, off" :: "v"(l), "v"(g) : "memory");
#endif
}

__device__ __forceinline__ void wait_async_zero() {
#if __has_builtin(__builtin_amdgcn_s_wait_asynccnt)
  __builtin_amdgcn_s_wait_asynccnt(0);
#else
  asm volatile("s_wait_asynccnt 0" ::: "memory");
#endif
  asm volatile("" ::: "memory");                 // keep LDS reads after the wait
}

// ---------- kernel 1: per-block (8-row) energies ----------

#define EAM_ROWBODY(TT)                                              \
  _Pragma("unroll")                                                  \
  for (int q = 0; q < 4; ++q) {                                      \
    const float d = dtile[r][jbase + 32 * q];                        \
    sp += fexp2(__builtin_fmaf(cPa[TT][q], d, cPb[TT][q]));          \
    sr += fexp2(__builtin_fmaf(cRa[TT][q], d, cRb[TT][q]));          \
  }

__global__ __launch_bounds__(256) void eam_rows(
    const int* __restrict__ types, const int* __restrict__ n_atoms,
    const float* __restrict__ dist,
    const float* __restrict__ phi_p, const float* __restrict__ rho_p,
    const float* __restrict__ emb_p, float* __restrict__ ws)
{
  __shared__ float dtile[kRowsPerBlock][kN];   // 32 KB distance tile
  __shared__ float sPa[6], sPb[6], sRa[6], sRb[6];
  __shared__ float pPhi[kRowsPerBlock][8];
  __shared__ float pRho[kRowsPerBlock][8];
  __shared__ float eRow[kRowsPerBlock];

  const int tid   = threadIdx.x;
  const int w     = tid >> 5;        // wave id (wave32)
  const int lane  = tid & 31;
  const int b     = blockIdx.y;
  const int ibase = blockIdx.x * kRowsPerBlock;
  const int nA    = n_atoms[b];

  // (1) kick off async staging: wave w copies distance row (ibase+w) into LDS
  {
    const float* grow = dist + ((size_t)(b * kN + ibase + w)) * kN;
    float* lrow = &dtile[w][0];
#pragma unroll
    for (int p = 0; p < 8; ++p)
      async_copy_b128(grow + p * 128 + lane * 4, lrow + p * 128 + lane * 4);
  }

  // (2) derived pair-type constants: phi = 2^(Pa*d+Pb), rho^2 = 2^(Ra*d+Rb)
  if (tid < 6) {
    float a = phi_p[tid * 3 + 0], bb = phi_p[tid * 3 + 1], c = phi_p[tid * 3 + 2];
    sPa[tid] = -bb * kLog2e;
    sPb[tid] = bb * c * kLog2e + __log2f(a);
    a = rho_p[tid * 3 + 0]; bb = rho_p[tid * 3 + 1]; c = rho_p[tid * 3 + 2];
    sRa[tid] = -2.0f * bb * kLog2e;
    sRb[tid] = 2.0f * (bb * c * kLog2e + __log2f(a));
  }
  __syncthreads();

  // (3) per-lane coefficient registers for its 4 columns x 3 possible row types.
  //     j >= n_atoms columns get (0, -12800) -> exp2 == 0: branch-free masking.
  const int jbase = w * 128 + lane;
  float cPa[3][4], cPb[3][4], cRa[3][4], cRb[3][4];
#pragma unroll
  for (int q = 0; q < 4; ++q) {
    const int j  = jbase + 32 * q;
    const int tj = types[b * kN + j];
    const bool act = (j < nA);
#pragma unroll
    for (int tt = 0; tt < 3; ++tt) {
      const int lo = tt < tj ? tt : tj;
      const int hi = tt < tj ? tj : tt;
      const int pt = lo * (7 - lo) / 2 + (hi - lo);   // lo*(2*NT-lo+1)/2 + (hi-lo), NT=3
      cPa[tt][q] = act ? sPa[pt] : 0.0f;
      cPb[tt][q] = act ? sPb[pt] : -12800.0f;
      cRa[tt][q] = act ? sRa[pt] : 0.0f;
      cRb[tt][q] = act ? sRb[pt] : -12800.0f;
    }
  }

  // (4) wait for async tile, publish to all waves
  wait_async_zero();
  __syncthreads();

  // (5) every wave processes its 128-column strip of all 8 rows:
  //     per pair: 1 ds_load_b32 + 2 v_fma + 2 v_exp + 2 v_add, no branches.
#pragma unroll
  for (int r = 0; r < kRowsPerBlock; ++r) {
    const int i = ibase + r;
    float sp = 0.0f, sr = 0.0f;
    if (i < nA) {
      const int ti = __builtin_amdgcn_readfirstlane(types[b * kN + i]); // uniform branch
      if (ti == 0)      { EAM_ROWBODY(0) }
      else if (ti == 1) { EAM_ROWBODY(1) }
      else              { EAM_ROWBODY(2) }
#pragma unroll
      for (int m = 16; m >= 1; m >>= 1) {   // wave32 butterfly reduce
        sp += __shfl_xor(sp, m, 32);
        sr += __shfl_xor(sr, m, 32);
      }
    }
    if (lane == 0) { pPhi[r][w] = sp; pRho[r][w] = sr; }
  }
  __syncthreads();

  // (6) finalize each row: combine 8 wave partials (fixed order), remove j==i
  //     self term, embedding, per-row energy.
  if (tid < kRowsPerBlock) {
    const int r = tid;
    const int i = ibase + r;
    float e = 0.0f;
    if (i < nA) {
      float sp = 0.0f, sr = 0.0f;
#pragma unroll
      for (int u = 0; u < 8; ++u) { sp += pPhi[r][u]; sr += pRho[r][u]; }
      const int ti  = types[b * kN + i];
      const float dii = dtile[r][i];
      const int pt  = ti * (7 - ti) / 2;            // pair type (ti, ti)
      sp -= fexp2(__builtin_fmaf(sPa[pt], dii, sPb[pt]));
      sr -= fexp2(__builtin_fmaf(sRa[pt], dii, sRb[pt]));
      sr  = fmaxf(sr, 1e-30f);
      e   = sp - emb_p[2 * ti] * sqrtf(sr) + emb_p[2 * ti + 1];
    }
    eRow[r] = e;
  }
  __syncthreads();
  if (tid == 0) {
    float s = 0.0f;
#pragma unroll
    for (int r2 = 0; r2 < kRowsPerBlock; ++r2) s += eRow[r2];
    ws[b * kBlocksX + blockIdx.x] = s;              // per-block partial energy
  }
}

// ---------- kernel 2: deterministic per-batch reduction + /n_atoms ----------

__global__ __launch_bounds__(128) void eam_reduce(
    const float* __restrict__ ws, const int* __restrict__ n_atoms,
    float* __restrict__ out)
{
  __shared__ float s[kBlocksX];
  const int b = blockIdx.x, tid = threadIdx.x;
  s[tid] = ws[b * kBlocksX + tid];
  __syncthreads();
#pragma unroll
  for (int st = kBlocksX / 2; st >= 1; st >>= 1) {
    if (tid < st) s[tid] += s[tid + st];
    __syncthreads();
  }
  if (tid == 0) out[b] = s[0] / (float)n_atoms[b];
}

// ---------- launch ----------

extern "C" void kernel_launch(void* const* d_in, const int* in_sizes, int n_in,
                              void* d_out, int out_size, void* d_ws, size_t ws_size,
                              hipStream_t stream) {
  const int*   types   = (const int*)  d_in[0];
  const int*   n_atoms = (const int*)  d_in[1];
  const float* dist    = (const float*)d_in[2];
  // d_in[3] (pair_types, 64 MB) intentionally unused: recomputed from types,
  // halving HBM traffic vs. the reference's input set.
  const float* phi_p   = (const float*)d_in[4];
  const float* rho_p   = (const float*)d_in[5];
  const float* emb_p   = (const float*)d_in[6];
  float* out = (float*)d_out;
  float* ws  = (float*)d_ws;   // needs kB * kBlocksX * 4 = 8 KB

  dim3 grid(kBlocksX, kB);
  hipLaunchKernelGGL(eam_rows, grid, dim3(256), 0, stream,
                     types, n_atoms, dist, phi_p, rho_p, emb_p, ws);
  hipLaunchKernelGGL(eam_reduce, dim3(kB), dim3(128), 0, stream,
                     ws, n_atoms, out);
}